// GIN_9972914061469
// MI455X (gfx1250) — compile-verified
//
#include <hip/hip_runtime.h>
#include <hip/hip_bf16.h>

typedef float v2f __attribute__((ext_vector_type(2)));
typedef float v8f __attribute__((ext_vector_type(8)));

#define DF 64  // feature width of node buffers (D_IN == D_HID == 64)

// ---------------------------------------------------------------------------
// Degree accumulation: one thread per edge, float atomics (resolve in L2).
__global__ void deg_kernel(const int* __restrict__ src, const int* __restrict__ dst,
                           float* __restrict__ deg_out, float* __restrict__ deg_in, int E) {
    int e = blockIdx.x * blockDim.x + threadIdx.x;
    if (e >= E) return;
    atomicAdd(&deg_out[src[e]], 1.0f);
    atomicAdd(&deg_in[dst[e]], 1.0f);
}

// In-place deg -> rsqrt(max(deg,1)) for both buffers.
__global__ void norm_kernel(float* __restrict__ a, float* __restrict__ b, int n) {
    int i = blockIdx.x * blockDim.x + threadIdx.x;
    if (i >= n) return;
    a[i] = __frsqrt_rn(fmaxf(a[i], 1.0f));
    b[i] = __frsqrt_rn(fmaxf(b[i], 1.0f));
}

// xs[node][:] = x[node][:] * s[node]; 16 threads per node, float4 each.
__global__ void prescale_kernel(const float* __restrict__ x, const float* __restrict__ s,
                                float* __restrict__ xs, int n_nodes) {
    int t = blockIdx.x * blockDim.x + threadIdx.x;
    int node = t >> 4;
    int q = t & 15;
    if (node >= n_nodes) return;
    float sc = s[node];
    float4 v = ((const float4*)(x + (size_t)node * DF))[q];
    v.x *= sc; v.y *= sc; v.z *= sc; v.w *= sc;
    ((float4*)(xs + (size_t)node * DF))[q] = v;
}

// agg[dst[e]][:] += xs[src[e]][:]; 16 threads per edge, float4 load + 4 atomics.
__global__ void scatter_kernel(const float* __restrict__ xs, const int* __restrict__ src,
                               const int* __restrict__ dst, float* __restrict__ agg, int E) {
    int t = blockIdx.x * blockDim.x + threadIdx.x;
    int e = t >> 4;
    int q = t & 15;
    if (e >= E) return;
    int s = src[e], d = dst[e];
    float4 v = ((const float4*)(xs + (size_t)s * DF))[q];
    float* p = agg + (size_t)d * DF + q * 4;
    atomicAdd(p + 0, v.x);
    atomicAdd(p + 1, v.y);
    atomicAdd(p + 2, v.z);
    atomicAdd(p + 3, v.w);
}

// Mean-pool numerator/denominator: 16 threads per node.
__global__ void pool_kernel(const float* __restrict__ h, const int* __restrict__ gid,
                            float* __restrict__ pool, float* __restrict__ cnt, int n_nodes) {
    int t = blockIdx.x * blockDim.x + threadIdx.x;
    int node = t >> 4;
    int q = t & 15;
    if (node >= n_nodes) return;
    int g = gid[node];
    float4 v = ((const float4*)(h + (size_t)node * DF))[q];
    float* p = pool + (size_t)g * DF + q * 4;
    atomicAdd(p + 0, v.x);
    atomicAdd(p + 1, v.y);
    atomicAdd(p + 2, v.z);
    atomicAdd(p + 3, v.w);
    if (q == 0) atomicAdd(&cnt[g], 1.0f);
}

__global__ void invcnt_kernel(float* __restrict__ cnt, int g) {
    int i = blockIdx.x * blockDim.x + threadIdx.x;
    if (i >= g) return;
    cnt[i] = 1.0f / fmaxf(cnt[i], 1.0f);
}

// ---------------------------------------------------------------------------
// out[r][c] = act( (A[r][:] * rowscale[r]) @ W + bias[c] )
// A: [rows, 64] (row-major, ld=64), W: [64, Ncols] row-major, out ld = Ncols.
// One 16x16 output tile per wave via V_WMMA_F32_16X16X4_F32, K=64 in 16 steps.
// blockDim.x = 32 * ceil(Ncols/16); wave w handles columns [16w, 16w+16).
//
// NCOLS_C > 0: compile-time Ncols (immediate offsets, no masking when 16|Ncols)
// NCOLS_C < 0: runtime Ncols from ncols_rt.
// B columns beyond Ncols are address-clamped, NOT zeroed: column n of B only
// influences column n of D, and out-of-range columns are never stored.
template <int NCOLS_C>
__global__ void gemm_wmma_kernel(const float* __restrict__ A,
                                 const float* __restrict__ rowscale,
                                 const float* __restrict__ W,
                                 const float* __restrict__ bias,
                                 float* __restrict__ out,
                                 int rows, int ncols_rt, int do_relu) {
    const int Ncols = (NCOLS_C > 0) ? NCOLS_C : ncols_rt;
    const int lane = threadIdx.x & 31;
    const int wave = threadIdx.x >> 5;
    const int half = lane >> 4;   // 0: lanes 0-15, 1: lanes 16-31
    const int r16  = lane & 15;
    const int m0 = blockIdx.x * 16;
    const int n0 = wave * 16;

    const int row  = m0 + r16;
    const int rowc = (row < rows) ? row : 0;   // clamp: duplicate row 0, never stored
    const float s = rowscale[rowc];
    // A layout for 16x16x4 f32 WMMA: lanes 0-15 supply K={4k,4k+1}, lanes 16-31 K={4k+2,4k+3}
    const float* arow = A + (size_t)rowc * DF + 2 * half;

    const int col  = n0 + r16;
    const bool cok = (col < Ncols);
    const int colc = cok ? col : (Ncols - 1);  // clamp address only; no zeroing needed
    const float* wp = W + (size_t)(2 * half) * Ncols + colc;

    v8f c = {};
#pragma unroll
    for (int k = 0; k < 16; ++k) {
        v2f a;
        a.x = arow[4 * k + 0] * s;
        a.y = arow[4 * k + 1] * s;
        v2f b;
        b.x = wp[(size_t)(4 * k) * Ncols];
        b.y = wp[(size_t)(4 * k + 1) * Ncols];
        // (neg_a, A, neg_b, B, c_mod, C, reuse_a, reuse_b)
        c = __builtin_amdgcn_wmma_f32_16x16x4_f32(false, a, false, b, (short)0, c,
                                                  false, false);
    }

    const float bv = bias[colc];
    float* orow0 = out + (size_t)m0 * Ncols + col;
    if (m0 + 16 <= rows && cok) {
        // Fast path: whole tile in range, unguarded stores with immediate offsets.
#pragma unroll
        for (int j = 0; j < 8; ++j) {
            float v = c[j] + bv;
            if (do_relu) v = fmaxf(v, 0.0f);
            orow0[(size_t)(j + 8 * half) * Ncols] = v;
        }
    } else if (cok) {
#pragma unroll
        for (int j = 0; j < 8; ++j) {
            int orow = m0 + j + 8 * half;
            if (orow < rows) {
                float v = c[j] + bv;
                if (do_relu) v = fmaxf(v, 0.0f);
                out[(size_t)orow * Ncols + col] = v;
            }
        }
    }
}

// ---------------------------------------------------------------------------
extern "C" void kernel_launch(void* const* d_in, const int* in_sizes, int n_in,
                              void* d_out, int out_size, void* d_ws, size_t ws_size,
                              hipStream_t stream) {
    const float* features = (const float*)d_in[0];   // [N, 64]
    const int*   src      = (const int*)d_in[1];     // [E]
    const int*   dst      = (const int*)d_in[2];     // [E]
    const int*   gid      = (const int*)d_in[3];     // [N]
    const float* W1       = (const float*)d_in[4];   // [64, 64]
    const float* b1       = (const float*)d_in[5];   // [64]
    const float* W2       = (const float*)d_in[6];   // [64, 64]
    const float* b2       = (const float*)d_in[7];   // [64]
    const float* Wc       = (const float*)d_in[8];   // [64, DOUT]
    const float* bc       = (const float*)d_in[9];   // [DOUT]
    float* outp           = (float*)d_out;

    const int N    = in_sizes[3];        // 100000 nodes
    const int E    = in_sizes[1];        // 1.6M edges
    const int DOUT = in_sizes[9];        // 8
    const int G    = out_size / DOUT;    // 512 graphs

    // Workspace layout (floats)
    float* ws    = (float*)d_ws;
    float* out_n = ws;                        // [N]   degrees -> out-norm (in place)
    float* in_n  = out_n + N;                 // [N]   degrees -> in-norm (in place)
    float* bufA  = in_n + N;                  // [N,64] prescaled features
    float* bufB  = bufA + (size_t)N * DF;     // [N,64] aggregated messages
    float* bufC  = bufB + (size_t)N * DF;     // [N,64] hidden activations
    float* pool  = bufC + (size_t)N * DF;     // [G,64]
    float* cnt   = pool + (size_t)G * DF;     // [G]

    const int TB = 256;

    // ---- degrees + normalization ----
    hipMemsetAsync(out_n, 0, sizeof(float) * (size_t)N * 2, stream);  // out_n + in_n
    deg_kernel<<<(E + TB - 1) / TB, TB, 0, stream>>>(src, dst, out_n, in_n, E);
    norm_kernel<<<(N + TB - 1) / TB, TB, 0, stream>>>(out_n, in_n, N);

    const int nodeT = N * 16;   // 16 threads per node
    const int edgeT = E * 16;   // 16 threads per edge
    const int gemmGrid = (N + 15) / 16;

    // ---- layer 1 ----
    prescale_kernel<<<(nodeT + TB - 1) / TB, TB, 0, stream>>>(features, out_n, bufA, N);
    hipMemsetAsync(bufB, 0, sizeof(float) * (size_t)N * DF, stream);
    scatter_kernel<<<(edgeT + TB - 1) / TB, TB, 0, stream>>>(bufA, src, dst, bufB, E);
    gemm_wmma_kernel<64><<<gemmGrid, 128, 0, stream>>>(bufB, in_n, W1, b1, bufC, N, DF, 1);

    // ---- layer 2 ----
    prescale_kernel<<<(nodeT + TB - 1) / TB, TB, 0, stream>>>(bufC, out_n, bufA, N);
    hipMemsetAsync(bufB, 0, sizeof(float) * (size_t)N * DF, stream);
    scatter_kernel<<<(edgeT + TB - 1) / TB, TB, 0, stream>>>(bufA, src, dst, bufB, E);
    gemm_wmma_kernel<64><<<gemmGrid, 128, 0, stream>>>(bufB, in_n, W2, b2, bufC, N, DF, 1);

    // ---- mean pool + classifier ----
    hipMemsetAsync(pool, 0, sizeof(float) * ((size_t)G * DF + G), stream);  // pool + cnt
    pool_kernel<<<(nodeT + TB - 1) / TB, TB, 0, stream>>>(bufC, gid, pool, cnt, N);
    invcnt_kernel<<<(G + TB - 1) / TB, TB, 0, stream>>>(cnt, G);

    const int clsWaves = (DOUT + 15) / 16;  // 1 wave (DOUT=8 fits in one 16-col tile)
    if (DOUT == 8) {
        gemm_wmma_kernel<8><<<(G + 15) / 16, 32 * clsWaves, 0, stream>>>(
            pool, cnt, Wc, bc, outp, G, DOUT, 0);
    } else {
        gemm_wmma_kernel<-1><<<(G + 15) / 16, 32 * clsWaves, 0, stream>>>(
            pool, cnt, Wc, bc, outp, G, DOUT, 0);
    }
}